// EMA_Layer_52853867545004
// MI455X (gfx1250) — compile-verified
//
#include <hip/hip_runtime.h>

// EMA linear recurrence e[t] = a*e[t-1] + alpha*x[t]  (e[0] = x[0]), a = 1-alpha = 0.1f
// over x of shape (B=8, T=4096, D=1024), fp32.
//
// Memory-bound (256 MiB traffic, ~11.5us at 23.3 TB/s). Strategy:
//  * chunk T into 512-step chunks; a = 0.1 means history older than 64 steps has
//    weight <= 1e-64 == exactly 0.0f in fp32, so a 64-step warm-up halo makes each
//    chunk independent (single pass, no inter-block carries, 12.5% extra reads).
//  * inside a chunk, process 16 timesteps x 16 features per wave as a matmul:
//        E(16x16) = M(16x16 decay matrix) * X(16x16) + a^(i+1)*carry
//    accumulated with 4 chained V_WMMA_F32_16X16X4_F32 (K=4 each).
//  * carry = row 15 of D = acc[7] in lanes 16..31 -> broadcast via __shfl.

typedef __attribute__((ext_vector_type(2))) float v2f;
typedef __attribute__((ext_vector_type(8))) float v8f;

#define ALPHA_F 0.9f

__device__ __forceinline__ float apown(float a, int n) {
    float p = 1.0f;
    for (int k = 0; k < n; ++k) p *= a;
    return p;
}

// M[i][j] = alpha * a^(i-j) for j<=i, else 0. First global tile: column 0 unscaled
// (e[0] = x[0] passes through with coefficient a^i, no alpha).
__device__ __forceinline__ float mcoef(float a, int i, int j, bool col0_unscaled) {
    if (j > i) return 0.0f;
    float v = apown(a, i - j);
    if (!(col0_unscaled && j == 0)) v *= ALPHA_F;
    return v;
}

__global__ __launch_bounds__(256) void ema_scan_wmma(const float* __restrict__ x,
                                                     float* __restrict__ out) {
    constexpr int T = 4096, D = 1024;
    constexpr int CHUNK = 512, HALO = 64;
    const float one_m = 1.0f - ALPHA_F;  // fp32 0.1 (matches reference arithmetic)

    const int lane = threadIdx.x & 31;
    const int half = lane >> 4;   // which 16-lane half of the wave
    const int ln   = lane & 15;   // row (A) / column-N (B,C,D) index
    const int wave = threadIdx.x >> 5;

    const int batch = blockIdx.z;
    const int chunk = blockIdx.y;
    const int feat  = blockIdx.x * 128 + wave * 16 + ln;

    // ---- constant A operand slices (16x4 each): lane=row, K = vgpr + 2*half ----
    const int jb = 2 * half;
    v2f A0n, A1s, A2s, A3s, A0f;
    A0n.x = mcoef(one_m, ln,      jb + 0, false);
    A0n.y = mcoef(one_m, ln,      jb + 1, false);
    A1s.x = mcoef(one_m, ln,  4 + jb + 0, false);
    A1s.y = mcoef(one_m, ln,  4 + jb + 1, false);
    A2s.x = mcoef(one_m, ln,  8 + jb + 0, false);
    A2s.y = mcoef(one_m, ln,  8 + jb + 1, false);
    A3s.x = mcoef(one_m, ln, 12 + jb + 0, false);
    A3s.y = mcoef(one_m, ln, 12 + jb + 1, false);
    A0f.x = mcoef(one_m, ln,      jb + 0, true);   // first-tile variant (col 0 unscaled)
    A0f.y = mcoef(one_m, ln,      jb + 1, true);

    // carry-injection factors: C/D VGPR r holds row (r + 8*half); seed = a^(row+1)*carry
    float cf[8];
#pragma unroll
    for (int r = 0; r < 8; ++r) cf[r] = apown(one_m, r + 1 + 8 * half);

    const bool firstchunk = (chunk == 0);
    const int nwarm  = firstchunk ? 0 : (HALO / 16);
    const int ntiles = nwarm + CHUNK / 16;
    int t = chunk * CHUNK - nwarm * 16;

    const float* xb = x   + (size_t)batch * T * D + feat;
    float*       ob = out + (size_t)batch * T * D + feat;

    const int kb = 2 * half;  // K index base for this half-wave in a 4-wide B slice
    float carry = 0.0f;

    for (int tile = 0; tile < ntiles; ++tile, t += 16) {
        // ---- load B slices (4x16 each): lane=feature column, K = vgpr + 2*half ----
        const float* p = xb + (size_t)t * D;
        v2f B0, B1, B2, B3;
        B0.x = p[(kb + 0)  * D];  B0.y = p[(kb + 1)  * D];
        B1.x = p[(kb + 4)  * D];  B1.y = p[(kb + 5)  * D];
        B2.x = p[(kb + 8)  * D];  B2.y = p[(kb + 9)  * D];
        B3.x = p[(kb + 12) * D];  B3.y = p[(kb + 13) * D];

        // seed accumulator with carry term: acc[r] = a^(row+1) * carry
        v8f acc;
#pragma unroll
        for (int r = 0; r < 8; ++r) acc[r] = cf[r] * carry;

        v2f a0 = (firstchunk && tile == 0) ? A0f : A0n;  // wave-uniform select
        acc = __builtin_amdgcn_wmma_f32_16x16x4_f32(false, a0,  false, B0, (short)0, acc, false, false);
        acc = __builtin_amdgcn_wmma_f32_16x16x4_f32(false, A1s, false, B1, (short)0, acc, false, false);
        acc = __builtin_amdgcn_wmma_f32_16x16x4_f32(false, A2s, false, B2, (short)0, acc, false, false);
        acc = __builtin_amdgcn_wmma_f32_16x16x4_f32(false, A3s, false, B3, (short)0, acc, false, false);

        // next-tile carry: row 15 lives in acc[7] of lanes 16..31 (feature = lane-16)
        carry = __shfl(acc[7], 16 + ln, 32);

        if (tile >= nwarm) {  // uniform branch: skip halo warm-up stores
            float* q = ob + (size_t)t * D;
#pragma unroll
            for (int r = 0; r < 8; ++r) q[(size_t)(r + 8 * half) * D] = acc[r];
        }
    }
}

extern "C" void kernel_launch(void* const* d_in, const int* in_sizes, int n_in,
                              void* d_out, int out_size, void* d_ws, size_t ws_size,
                              hipStream_t stream) {
    (void)in_sizes; (void)n_in; (void)out_size; (void)d_ws; (void)ws_size;
    const float* x = (const float*)d_in[0];
    float* out = (float*)d_out;
    // grid: (feature groups of 128, T chunks of 512, batches) = (8, 8, 8) -> 512 blocks, 4096 waves
    dim3 block(256);
    dim3 grid(1024 / 128, 4096 / 512, 8);
    ema_scan_wmma<<<grid, block, 0, stream>>>(x, out);
}